// GCN_18056042512717
// MI455X (gfx1250) — compile-verified
//
#include <hip/hip_runtime.h>
#include <hip/hip_bf16.h>

#define GCN_N_NODES 100000

typedef float v2f __attribute__((ext_vector_type(2)));
typedef float v8f __attribute__((ext_vector_type(8)));

// ---------------------------------------------------------------------------
// Dense transform  H[M x Nout] = X[M x K] @ W[K x Nout]  via V_WMMA_F32_16X16X4_F32.
// One wave32 per 16-row tile; wave computes all NT (= Nout/16) column tiles so X
// is read exactly once. M = 100000 = 6250 * 16 (no tail).
// A layout (f32 16x4): lane L holds A[L&15][k + 2*(L>>4) + v] in VGPR v.
// B layout (f32 4x16): lane L holds B[k + 2*(L>>4) + v][L&15] in VGPR v.
// C/D layout: c[r] = C[r + 8*(L>>4)][L&15].
// ---------------------------------------------------------------------------
template <int K, int NT>
__global__ __launch_bounds__(256) void gcn_gemm_wmma_f32(
    const float* __restrict__ X, const float* __restrict__ W,
    float* __restrict__ H, int mTiles) {
  const int lane    = threadIdx.x & 31;
  const int wave    = threadIdx.x >> 5;
  const int rowTile = blockIdx.x * (blockDim.x >> 5) + wave;
  if (rowTile >= mTiles) return;  // wave-uniform: EXEC all-1s for WMMA

  const int m15  = lane & 15;
  const int kh   = lane >> 4;  // 0 or 1
  const int Nout = NT * 16;
  const float* __restrict__ xrow = X + (size_t)(rowTile * 16 + m15) * K;

  v8f acc[NT];
#pragma unroll
  for (int t = 0; t < NT; ++t) acc[t] = v8f{0.f, 0.f, 0.f, 0.f, 0.f, 0.f, 0.f, 0.f};

#pragma unroll 4
  for (int k = 0; k < K; k += 4) {
    v2f a;
    a.x = xrow[k + 2 * kh + 0];
    a.y = xrow[k + 2 * kh + 1];
#pragma unroll
    for (int t = 0; t < NT; ++t) {
      v2f b;
      b.x = W[(k + 2 * kh + 0) * Nout + t * 16 + m15];
      b.y = W[(k + 2 * kh + 1) * Nout + t * 16 + m15];
      // 8 args: (neg_a, A, neg_b, B, c_mod, C, reuse_a, reuse_b)
      acc[t] = __builtin_amdgcn_wmma_f32_16x16x4_f32(
          false, a, false, b, (short)0, acc[t], false, false);
    }
  }

#pragma unroll
  for (int t = 0; t < NT; ++t)
#pragma unroll
    for (int r = 0; r < 8; ++r)
      H[(size_t)(rowTile * 16 + r + 8 * kh) * Nout + t * 16 + m15] = acc[t][r];
}

// ---------------------------------------------------------------------------
// deg[d] += 1 over edges (atomics resolve in L2; deg array is only 400 KB).
// ---------------------------------------------------------------------------
__global__ void gcn_degree_kernel(const int* __restrict__ dst,
                                  float* __restrict__ deg, int E) {
  int e = blockIdx.x * blockDim.x + threadIdx.x;
  if (e < E) atomicAdd(&deg[dst[e]], 1.0f);
}

// dinv[i] = rsqrt(deg[i] + 1)  (in place; +1 = self loop)
__global__ void gcn_dinv_kernel(float* __restrict__ deg, int n) {
  int i = blockIdx.x * blockDim.x + threadIdx.x;
  if (i < n) deg[i] = __frsqrt_rn(deg[i] + 1.0f);
}

// ---------------------------------------------------------------------------
// Edge aggregation: agg[dst] += h[src] * dinv[src]*dinv[dst].
// F threads per edge (F=64: 2 waves/edge, F=32: 1 wave/edge); per-edge scalars
// are wave-uniform loads, the feature gather/scatter is fully coalesced and
// stays resident in the 192 MB L2 (h and agg are 25.6 MB each).
// ---------------------------------------------------------------------------
template <int F>
__global__ __launch_bounds__(256) void gcn_aggregate_kernel(
    const float* __restrict__ h, const int* __restrict__ src,
    const int* __restrict__ dst, const float* __restrict__ dinv,
    float* __restrict__ agg, int E) {
  constexpr int EPB = 256 / F;
  const int e = blockIdx.x * EPB + (threadIdx.x / F);
  if (e >= E) return;
  const int f = threadIdx.x % F;
  const int s = src[e];
  const int d = dst[e];
  const float norm = dinv[s] * dinv[d];
  atomicAdd(&agg[(size_t)d * F + f], h[(size_t)s * F + f] * norm);
}

// ---------------------------------------------------------------------------
// Self-loop + bias (+ optional ReLU), in place on agg.
// out = relu?(agg + h * dinv^2 + b)
// ---------------------------------------------------------------------------
template <int F, bool RELU>
__global__ void gcn_finish_kernel(float* __restrict__ agg,
                                  const float* __restrict__ h,
                                  const float* __restrict__ dinv,
                                  const float* __restrict__ bias, int n) {
  int idx = blockIdx.x * blockDim.x + threadIdx.x;
  if (idx >= n * F) return;
  const int i = idx / F;
  const int f = idx % F;
  const float di = dinv[i];
  float v = agg[idx] + h[idx] * di * di + bias[f];
  if (RELU) v = fmaxf(v, 0.0f);
  agg[idx] = v;
}

// ---------------------------------------------------------------------------
extern "C" void kernel_launch(void* const* d_in, const int* in_sizes, int n_in,
                              void* d_out, int out_size, void* d_ws, size_t ws_size,
                              hipStream_t stream) {
  const float* x  = (const float*)d_in[0];
  const int*   ei = (const int*)d_in[1];
  const float* W1 = (const float*)d_in[2];
  const float* b1 = (const float*)d_in[3];
  const float* W2 = (const float*)d_in[4];
  const float* b2 = (const float*)d_in[5];

  const int N = GCN_N_NODES;
  const int E = in_sizes[1] / 2;
  const int* src = ei;
  const int* dst = ei + E;

  // Workspace carve-up (all offsets 16B-aligned).
  char* ws = (char*)d_ws;
  float* dinv = (float*)ws;                              // N floats (deg -> dinv in place)
  float* h1   = (float*)(ws + (size_t)N * sizeof(float));        // N x 64
  float* agg1 = h1 + (size_t)N * 64;                              // N x 64
  float* h2   = agg1 + (size_t)N * 64;                            // N x 32
  float* outp = (float*)d_out;                                    // N x 32

  // ---- shared normalization (edge list identical for both layers) ----
  hipMemsetAsync(dinv, 0, (size_t)N * sizeof(float), stream);
  gcn_degree_kernel<<<(E + 255) / 256, 256, 0, stream>>>(dst, dinv, E);
  gcn_dinv_kernel<<<(N + 255) / 256, 256, 0, stream>>>(dinv, N);

  // ---- layer 1: h1 = x @ W1 ; agg1 = norm-agg(h1) ; relu(+b1) in place ----
  const int mTiles = N / 16;  // 6250, exact
  gcn_gemm_wmma_f32<128, 4><<<(mTiles + 7) / 8, 256, 0, stream>>>(x, W1, h1, mTiles);
  hipMemsetAsync(agg1, 0, (size_t)N * 64 * sizeof(float), stream);
  gcn_aggregate_kernel<64><<<(E + 3) / 4, 256, 0, stream>>>(h1, src, dst, dinv, agg1, E);
  gcn_finish_kernel<64, true><<<(N * 64 + 255) / 256, 256, 0, stream>>>(agg1, h1, dinv, b1, N);

  // ---- layer 2: h2 = agg1 @ W2 ; d_out = norm-agg(h2) ; +b2 in place ----
  gcn_gemm_wmma_f32<64, 2><<<(mTiles + 7) / 8, 256, 0, stream>>>(agg1, W2, h2, mTiles);
  hipMemsetAsync(outp, 0, (size_t)N * 32 * sizeof(float), stream);
  gcn_aggregate_kernel<32><<<(E + 7) / 8, 256, 0, stream>>>(h2, src, dst, dinv, outp, E);
  gcn_finish_kernel<32, false><<<(N * 32 + 255) / 256, 256, 0, stream>>>(outp, h2, dinv, b2, N);
}